// scaled_hessian_RBF2_10058813407289
// MI455X (gfx1250) — compile-verified
//
#include <hip/hip_runtime.h>
#include <hip/hip_bf16.h>

// N = 8192, D = 128, all fp32.
// out = [ K (N*N) | dK (N*D) ] concatenated, fp32.

#define NN 8192
#define DD 128
#define JSPLIT 8

typedef float v2f __attribute__((ext_vector_type(2)));
typedef float v4f __attribute__((ext_vector_type(4)));
typedef float v8f __attribute__((ext_vector_type(8)));

__device__ __forceinline__ v8f wmma_f32_4(v2f a, v2f b, v8f c) {
  // D = A(16x4) * B(4x16) + C(16x16), fp32, wave32
  return __builtin_amdgcn_wmma_f32_16x16x4_f32(
      /*neg_a=*/false, a, /*neg_b=*/false, b,
      /*c_mod=*/(short)0, c, /*reuse_a=*/false, /*reuse_b=*/false);
}

// ---------------------------------------------------------------------------
// Pass 1: XS = X @ S (N x D), qX[i] = sum_d X[i,d]*XS[i,d]; optionally XS^T.
// ---------------------------------------------------------------------------
__global__ void prep_kernel(const float* __restrict__ Xin,
                            const float* __restrict__ S,
                            float* __restrict__ XSo,
                            float* __restrict__ qXo,
                            float* __restrict__ XSTo) {
  extern __shared__ float sm[];
  float* smS = sm;                    // 128 x 132
  float* smX = smS + 128 * 132;       // 16 x 132
  float* smQ = smX + 16 * 132;        // 16 x 128
  const int t = threadIdx.x;
  const int r0 = blockIdx.x * 16;

  for (int it = 0; it < 16; ++it) {
    int q = t + it * 256;
    int row = q >> 5, c4 = (q & 31) << 2;
    *(v4f*)(smS + row * 132 + c4) = *(const v4f*)(S + row * DD + c4);
  }
  for (int it = 0; it < 2; ++it) {
    int q = t + it * 256;
    int row = q >> 5, c4 = (q & 31) << 2;
    *(v4f*)(smX + row * 132 + c4) =
        *(const v4f*)(Xin + (size_t)(r0 + row) * DD + c4);
  }
  __syncthreads();

  const int d = t & 127, g = t >> 7;
  for (int rr = 0; rr < 8; ++rr) {
    int row = g * 8 + rr;
    float a = 0.f;
    for (int k = 0; k < DD; ++k) a = fmaf(smX[row * 132 + k], smS[k * 132 + d], a);
    XSo[(size_t)(r0 + row) * DD + d] = a;
    if (XSTo) XSTo[(size_t)d * NN + r0 + row] = a;  // transposed copy
    smQ[row * 128 + d] = smX[row * 132 + d] * a;
  }
  __syncthreads();
  for (int s = 64; s > 0; s >>= 1) {
    if (d < s) {
      for (int rr = 0; rr < 8; ++rr) {
        int row = g * 8 + rr;
        smQ[row * 128 + d] += smQ[row * 128 + d + s];
      }
    }
    __syncthreads();
  }
  if (d == 0)
    for (int rr = 0; rr < 8; ++rr) {
      int row = g * 8 + rr;
      qXo[r0 + row] = smQ[row * 128];
    }
}

// ---------------------------------------------------------------------------
// Pass 2: K[i,j] = exp(-0.5 * (qY[i] + qX[j] - YS[i].X[j] - Y[i].XS[j]))
// Workgroup tile 128x128, wave tile 32x64. Frag-swizzled LDS (k = 4s+2hi+e
// at pos hi*64+2s+e): one b128 per fragment covers two k-steps; 6 b128 feed
// 16 WMMAs per iteration; next iteration prefetched under the WMMA chain.
// ---------------------------------------------------------------------------
__global__ void kxy_kernel(const float* __restrict__ X,
                           const float* __restrict__ Y,
                           const float* __restrict__ XS,
                           const float* __restrict__ YS,
                           const float* __restrict__ qX,
                           const float* __restrict__ qY,
                           float* __restrict__ Kout) {
  extern __shared__ float sm[];
  float* smA = sm;              // 128 x 132 (i rows, swizzled k)
  float* smB = sm + 128 * 132;  // 128 x 132 (j rows, swizzled k)
  const int t = threadIdx.x;
  const int lane = t & 31, w = t >> 5;
  const int n = lane & 15, hi = lane >> 4;
  const int iw = (w & 3) * 32;
  const int jw = (w >> 2) * 64;
  const int i0 = blockIdx.x * 128, j0 = blockIdx.y * 128;

  v8f acc[2][4];
#pragma unroll
  for (int a = 0; a < 2; ++a)
#pragma unroll
    for (int b = 0; b < 4; ++b)
      acc[a][b] = (v8f){0.f, 0.f, 0.f, 0.f, 0.f, 0.f, 0.f, 0.f};

  const float* aPtr0 = smA + (iw + n) * 132 + hi * 64;
  const float* aPtr1 = smA + (iw + 16 + n) * 132 + hi * 64;
  const float* bPtr0 = smB + (jw + n) * 132 + hi * 64;
  const float* bPtr1 = smB + (jw + 16 + n) * 132 + hi * 64;
  const float* bPtr2 = smB + (jw + 32 + n) * 132 + hi * 64;
  const float* bPtr3 = smB + (jw + 48 + n) * 132 + hi * 64;

  for (int kb = 0; kb < 2; ++kb) {
    const float* Asrc = kb ? Y : YS;
    const float* Bsrc = kb ? XS : X;
    __syncthreads();
    for (int it = 0; it < 16; ++it) {
      int q = t + it * 256;
      int row = q >> 5, c4 = (q & 31) << 2;
      v4f va = *(const v4f*)(Asrc + (size_t)(i0 + row) * DD + c4);
      v4f vb = *(const v4f*)(Bsrc + (size_t)(j0 + row) * DD + c4);
      *(v2f*)(smA + row * 132 + (c4 >> 1)) = va.xy;
      *(v2f*)(smA + row * 132 + 64 + (c4 >> 1)) = va.zw;
      *(v2f*)(smB + row * 132 + (c4 >> 1)) = vb.xy;
      *(v2f*)(smB + row * 132 + 64 + (c4 >> 1)) = vb.zw;
    }
    __syncthreads();

    v4f a0 = *(const v4f*)(aPtr0);
    v4f a1 = *(const v4f*)(aPtr1);
    v4f b0 = *(const v4f*)(bPtr0);
    v4f b1 = *(const v4f*)(bPtr1);
    v4f b2 = *(const v4f*)(bPtr2);
    v4f b3 = *(const v4f*)(bPtr3);
#pragma unroll 4
    for (int it = 0; it < 16; ++it) {
      const int off = ((it + 1) & 15) << 2;
      v4f a0n = *(const v4f*)(aPtr0 + off);
      v4f a1n = *(const v4f*)(aPtr1 + off);
      v4f b0n = *(const v4f*)(bPtr0 + off);
      v4f b1n = *(const v4f*)(bPtr1 + off);
      v4f b2n = *(const v4f*)(bPtr2 + off);
      v4f b3n = *(const v4f*)(bPtr3 + off);

      acc[0][0] = wmma_f32_4(a0.xy, b0.xy, acc[0][0]);
      acc[0][1] = wmma_f32_4(a0.xy, b1.xy, acc[0][1]);
      acc[0][2] = wmma_f32_4(a0.xy, b2.xy, acc[0][2]);
      acc[0][3] = wmma_f32_4(a0.xy, b3.xy, acc[0][3]);
      acc[1][0] = wmma_f32_4(a1.xy, b0.xy, acc[1][0]);
      acc[1][1] = wmma_f32_4(a1.xy, b1.xy, acc[1][1]);
      acc[1][2] = wmma_f32_4(a1.xy, b2.xy, acc[1][2]);
      acc[1][3] = wmma_f32_4(a1.xy, b3.xy, acc[1][3]);
      acc[0][0] = wmma_f32_4(a0.zw, b0.zw, acc[0][0]);
      acc[0][1] = wmma_f32_4(a0.zw, b1.zw, acc[0][1]);
      acc[0][2] = wmma_f32_4(a0.zw, b2.zw, acc[0][2]);
      acc[0][3] = wmma_f32_4(a0.zw, b3.zw, acc[0][3]);
      acc[1][0] = wmma_f32_4(a1.zw, b0.zw, acc[1][0]);
      acc[1][1] = wmma_f32_4(a1.zw, b1.zw, acc[1][1]);
      acc[1][2] = wmma_f32_4(a1.zw, b2.zw, acc[1][2]);
      acc[1][3] = wmma_f32_4(a1.zw, b3.zw, acc[1][3]);

      a0 = a0n; a1 = a1n;
      b0 = b0n; b1 = b1n; b2 = b2n; b3 = b3n;
    }
  }

  float qy[2][8];
#pragma unroll
  for (int ii = 0; ii < 2; ++ii)
#pragma unroll
    for (int r = 0; r < 8; ++r)
      qy[ii][r] = qY[i0 + iw + ii * 16 + hi * 8 + r];
#pragma unroll
  for (int ii = 0; ii < 2; ++ii) {
#pragma unroll
    for (int tj = 0; tj < 4; ++tj) {
      float qx = qX[j0 + jw + tj * 16 + n];
#pragma unroll
      for (int r = 0; r < 8; ++r) {
        int m = hi * 8 + r;
        float quad = qy[ii][r] + qx - acc[ii][tj][r];
        Kout[(size_t)(i0 + iw + ii * 16 + m) * NN + (j0 + jw + tj * 16 + n)] =
            __expf(-0.5f * quad);
      }
    }
  }
}

// ---------------------------------------------------------------------------
// Pass 3: partials of K @ XS and rowsum(K) (WMMA vs ones B-matrix).
// B side uses XS^T so both A (K tile) and B (XS^T tile) use the swizzled
// b128 frag layout (k = 4s+2hi+e at pos hi*32+2s+e, stride 68):
// 9 ds_load_b128 feed 18 WMMAs per 2-k-step iteration, pipelined.
// ---------------------------------------------------------------------------
__global__ void dk_partial_kernel(const float* __restrict__ Kmat,
                                  const float* __restrict__ XST,
                                  float* __restrict__ part,
                                  float* __restrict__ partRS) {
  extern __shared__ float sm[];
  float* smXT = sm;              // 128 x 68 (d rows, swizzled j)
  float* smK = sm + 128 * 68;    // 8 waves x (16 x 68)
  const int t = threadIdx.x, lane = t & 31, w = t >> 5;
  const int n = lane & 15, hi = lane >> 4;
  const int i0 = blockIdx.x * 128 + w * 16;
  const int jc0 = blockIdx.y * (NN / JSPLIT);
  float* smKw = smK + w * 16 * 68;

  v8f accd[8];
#pragma unroll
  for (int q = 0; q < 8; ++q)
    accd[q] = (v8f){0.f, 0.f, 0.f, 0.f, 0.f, 0.f, 0.f, 0.f};
  v8f accs = (v8f){0.f, 0.f, 0.f, 0.f, 0.f, 0.f, 0.f, 0.f};
  const v2f ones = {1.f, 1.f};

  const float* aP = smKw + n * 68 + hi * 32;
  const float* bP[8];
#pragma unroll
  for (int td = 0; td < 8; ++td) bP[td] = smXT + (td * 16 + n) * 68 + hi * 32;

  for (int jb = jc0; jb < jc0 + NN / JSPLIT; jb += 64) {
    __syncthreads();
    // stage XS^T rows d=0..127, j-cols jb..jb+63 (swizzled)
    for (int it = 0; it < 8; ++it) {
      int q = t + it * 256;                    // 2048 float4
      int row = q >> 4, c4 = (q & 15) << 2;    // 128 rows x 16 groups
      v4f v = *(const v4f*)(XST + (size_t)row * NN + jb + c4);
      *(v2f*)(smXT + row * 68 + (c4 >> 1)) = v.xy;
      *(v2f*)(smXT + row * 68 + 32 + (c4 >> 1)) = v.zw;
    }
    // stage this wave's 16x64 K tile (swizzled)
    for (int rr = 0; rr < 8; ++rr) {
      int row = rr * 2 + hi;
      int c = n << 2;
      v4f v = *(const v4f*)(Kmat + (size_t)(i0 + row) * NN + jb + c);
      *(v2f*)(smKw + row * 68 + (c >> 1)) = v.xy;
      *(v2f*)(smKw + row * 68 + 32 + (c >> 1)) = v.zw;
    }
    __syncthreads();

    v4f a = *(const v4f*)(aP);
    v4f b[8];
#pragma unroll
    for (int td = 0; td < 8; ++td) b[td] = *(const v4f*)(bP[td]);
#pragma unroll 2
    for (int it = 0; it < 8; ++it) {
      const int off = ((it + 1) & 7) << 2;  // wrap: harmless reload on last
      v4f an = *(const v4f*)(aP + off);
      v4f bn[8];
#pragma unroll
      for (int td = 0; td < 8; ++td) bn[td] = *(const v4f*)(bP[td] + off);

      accs = wmma_f32_4(a.xy, ones, accs);
#pragma unroll
      for (int td = 0; td < 8; ++td)
        accd[td] = wmma_f32_4(a.xy, b[td].xy, accd[td]);
      accs = wmma_f32_4(a.zw, ones, accs);
#pragma unroll
      for (int td = 0; td < 8; ++td)
        accd[td] = wmma_f32_4(a.zw, b[td].zw, accd[td]);

      a = an;
#pragma unroll
      for (int td = 0; td < 8; ++td) b[td] = bn[td];
    }
  }

  const int c = blockIdx.y;
#pragma unroll
  for (int td = 0; td < 8; ++td)
#pragma unroll
    for (int r = 0; r < 8; ++r) {
      int m = hi * 8 + r;
      part[(size_t)c * NN * DD + (size_t)(i0 + m) * DD + td * 16 + n] =
          accd[td][r];
    }
  if (n == 0)
#pragma unroll
    for (int r = 0; r < 8; ++r) {
      int m = hi * 8 + r;
      partRS[c * NN + i0 + m] = accs[r];
    }
}

// ---------------------------------------------------------------------------
// Pass 4: dK[i,d] = 2 * (rowsum(K)[i] * YS[i,d] - sum_c part[c,i,d])
// ---------------------------------------------------------------------------
__global__ void reduce_kernel(const float* __restrict__ part,
                              const float* __restrict__ partRS,
                              const float* __restrict__ YS,
                              float* __restrict__ dK) {
  int idx = blockIdx.x * 256 + threadIdx.x;
  int i = idx >> 7;
  float s = 0.f, rs = 0.f;
#pragma unroll
  for (int c = 0; c < JSPLIT; ++c) {
    s += part[(size_t)c * NN * DD + idx];
    rs += partRS[c * NN + i];
  }
  dK[idx] = 2.f * (rs * YS[idx] - s);
}

// ---------------------------------------------------------------------------
extern "C" void kernel_launch(void* const* d_in, const int* in_sizes, int n_in,
                              void* d_out, int out_size, void* d_ws,
                              size_t ws_size, hipStream_t stream) {
  const float* X = (const float*)d_in[0];
  const float* Y = (const float*)d_in[1];
  const float* S = (const float*)d_in[2];

  float* Kout = (float*)d_out;                  // N*N
  float* dK = (float*)d_out + (size_t)NN * NN;  // N*D

  // ws layout (floats): XS | YS | qX | qY | part | partRS | XST
  float* ws = (float*)d_ws;
  float* XS = ws;
  float* YS = XS + (size_t)NN * DD;
  float* qX = YS + (size_t)NN * DD;
  float* qY = qX + NN;
  float* part = qY + NN;
  float* partRS = part + (size_t)JSPLIT * NN * DD;
  float* XST = partRS + (size_t)JSPLIT * NN;

  const size_t smPrep = (size_t)(128 * 132 + 16 * 132 + 16 * 128) * sizeof(float);
  prep_kernel<<<dim3(NN / 16), 256, smPrep, stream>>>(X, S, XS, qX, XST);
  prep_kernel<<<dim3(NN / 16), 256, smPrep, stream>>>(Y, S, YS, qY, nullptr);

  const size_t smKxy = (size_t)(2 * 128 * 132) * sizeof(float);
  kxy_kernel<<<dim3(NN / 128, NN / 128), 256, smKxy, stream>>>(X, Y, XS, YS, qX,
                                                               qY, Kout);

  const size_t smDk = (size_t)(128 * 68 + 8 * 16 * 68) * sizeof(float);
  dk_partial_kernel<<<dim3(NN / 128, JSPLIT), 256, smDk, stream>>>(Kout, XST,
                                                                   part, partRS);

  reduce_kernel<<<dim3((NN * DD) / 256), 256, 0, stream>>>(part, partRS, YS, dK);
}